// MultivariatePowerFrequencyFeatures_74818330296541
// MI455X (gfx1250) — compile-verified
//
#include <hip/hip_runtime.h>
#include <hip/hip_bf16.h>

typedef float v2f __attribute__((ext_vector_type(2)));
typedef float v8f __attribute__((ext_vector_type(8)));

#define NW   308      // number of sliding windows
#define NB   16       // batch
#define NC   16       // channels
#define NT   20000    // samples
#define SPITCH 18     // padded pitch for spec tiles (cols per row)

static __device__ __forceinline__ v8f wmma4(v2f a, v2f b, v8f c) {
    // D = A(16x4,f32) * B(4x16,f32) + C(16x16,f32); f32 WMMA: no A/B neg allowed.
    return __builtin_amdgcn_wmma_f32_16x16x4_f32(false, a, false, b, (short)0, c, false, false);
}

// ---------------------------------------------------------------------------
// Kernel A: fused STFT (WMMA f32 16x16x4) + CSD (WMMA) + |.|^2 + band pooling.
// One workgroup (128 threads = 4 waves) per (b, n).
// Writes pooled[b][n][k][c][d] (k = 5 bands) to workspace, pre-scaled.
// ---------------------------------------------------------------------------
__global__ __launch_bounds__(128)
void mpff_stft_csd_kernel(const float* __restrict__ x,
                          const float* __restrict__ win,
                          float* __restrict__ pooled)
{
    __shared__ float lds_xin[NC * 304];     // 16 ch x 304 samples
    __shared__ float lds_sre[256 * SPITCH]; // spec real, rows = c*16+j, 16 freq cols
    __shared__ float lds_sim[256 * SPITCH]; // spec imag
    __shared__ float lds_pool[5 * 256];     // band accumulators
    __shared__ float lds_win[64];

    const int tid  = threadIdx.x;
    const int lane = tid & 31;
    const int wv   = tid >> 5;            // wave 0..3
    const int blk  = blockIdx.x;          // 0..16*308-1
    const int b    = blk / NW;
    const int n    = blk % NW;
    const int l15  = lane & 15;
    const int lhalf = (lane < 16) ? 0 : 2;  // K-offset for this lane half
    const int mofs  = (lane < 16) ? 0 : 8;  // C/D row offset for this lane half

    // zero band pool
    for (int i = tid; i < 5 * 256; i += 128) lds_pool[i] = 0.0f;
    // window to LDS
    if (tid < 64) lds_win[tid] = win[tid];
    // load 16 channels x 304 samples (always in-bounds: max end = 19951 < 20000)
    const float* xb = x + (size_t)b * (NC * NT) + (size_t)n * 64;
    for (int i = tid; i < NC * 304; i += 128) {
        int c = i / 304, t = i % 304;
        lds_xin[c * 304 + t] = xb[(size_t)c * NT + t];
    }
    __syncthreads();

    // 1/||window||
    float s2 = 0.0f;
    #pragma unroll
    for (int t = 0; t < 64; ++t) { float w = lds_win[t]; s2 += w * w; }
    const float inorm = 1.0f / sqrtf(s2);

    // process freq bins in 3 tiles of 16: [0..15],[16..31],[32(+junk)]
    for (int pr = 0; pr < 3; ++pr) {
        const int fbase = pr * 16;

        // ---- STFT phase ----
        // wave -> (part: 0=re,1=im ; channel group)
        const int part = wv & 1;
        const int cgrp = wv >> 1;

        // Per-lane B operand values for all 16 K-steps (DFT matrix column fbase+l15).
        // B[k=t][n=f]: lane l15 = f-col; v0: t = 4s+lhalf, v1: t+1.
        const int fcol = fbase + l15;
        float Bv0[16], Bv1[16];
        #pragma unroll
        for (int s = 0; s < 16; ++s) {
            int t0 = 4 * s + lhalf;
            int t1 = t0 + 1;
            float th0 = (float)((t0 * fcol) & 63) * 0.0981747704f;  // pi/32
            float th1 = (float)((t1 * fcol) & 63) * 0.0981747704f;
            float w0 = lds_win[t0] * inorm;
            float w1 = lds_win[t1] * inorm;
            if (part == 0) { Bv0[s] =  w0 * __cosf(th0); Bv1[s] =  w1 * __cosf(th1); }
            else           { Bv0[s] = -w0 * __sinf(th0); Bv1[s] = -w1 * __sinf(th1); }
        }

        for (int ci = 0; ci < 8; ++ci) {
            const int c = cgrp * 8 + ci;
            const float* xc = &lds_xin[c * 304];
            v8f acc = {0.f,0.f,0.f,0.f,0.f,0.f,0.f,0.f};
            #pragma unroll
            for (int s = 0; s < 16; ++s) {
                int t0 = 4 * s + lhalf;
                v2f a, bb;
                a[0]  = xc[l15 * 16 + t0];       // frame j=l15, sample t0
                a[1]  = xc[l15 * 16 + t0 + 1];
                bb[0] = Bv0[s];
                bb[1] = Bv1[s];
                acc = wmma4(a, bb, acc);
            }
            float* dst = (part == 0) ? lds_sre : lds_sim;
            #pragma unroll
            for (int rr = 0; rr < 8; ++rr) {
                int j = rr + mofs;               // row-in-tile (frame index)
                dst[(c * 16 + j) * SPITCH + l15] = acc[rr];
            }
        }
        __syncthreads();

        // ---- CSD + pooling phase ----
        // freq list per wave (f=0 excluded; bands cover f=1..32 only)
        int fstart, fcnt;
        if (pr == 0)      { fstart = (wv == 0) ? 4 : wv; fcnt = (wv == 0) ? 3 : 4; }
        else if (pr == 1) { fstart = 16 + wv;            fcnt = 4; }
        else              { fstart = 32;                 fcnt = (wv == 0) ? 1 : 0; }

        for (int fi = 0; fi < fcnt; ++fi) {
            const int f    = fstart + 4 * fi;
            const int col  = f - fbase;
            const int band = (f < 2) ? 0 : (f < 4) ? 1 : (f < 7) ? 2 : (f < 13) ? 3 : 4;

            v8f P = {0.f,0.f,0.f,0.f,0.f,0.f,0.f,0.f};
            v8f Q = {0.f,0.f,0.f,0.f,0.f,0.f,0.f,0.f};
            #pragma unroll
            for (int s = 0; s < 4; ++s) {
                int j0   = 4 * s + lhalf;
                int row0 = l15 * 16 + j0;        // channel=l15, frame=j0
                v2f are, aim, nre;
                are[0] = lds_sre[row0 * SPITCH + col];
                are[1] = lds_sre[(row0 + 1) * SPITCH + col];
                aim[0] = lds_sim[row0 * SPITCH + col];
                aim[1] = lds_sim[(row0 + 1) * SPITCH + col];
                nre[0] = -are[0];
                nre[1] = -are[1];
                // same registers serve A (16x4, M=c) and B (4x16, N=d) for S*S^T
                P = wmma4(are, are, P);          // Re Re^T
                P = wmma4(aim, aim, P);          // + Im Im^T
                Q = wmma4(aim, are, Q);          // Im Re^T
                Q = wmma4(nre, aim, Q);          // - Re Im^T
            }
            #pragma unroll
            for (int rr = 0; rr < 8; ++rr) {
                float v = P[rr] * P[rr] + Q[rr] * Q[rr];   // |csd*16|^2
                int i = rr + mofs;
                atomicAdd(&lds_pool[band * 256 + i * 16 + l15], v);  // ds_add_f32
            }
        }
        __syncthreads();  // spec buffers reused next tile; pool complete at end
    }

    // scale = 1/(w^2 * denom[band]); w=16
    const float sc[5] = {1.0f/256.0f, 1.0f/512.0f, 1.0f/768.0f, 1.0f/1536.0f, 1.0f/5120.0f};
    float* dst = pooled + ((size_t)(b * NW + n)) * (5 * 256);
    for (int i = tid; i < 5 * 256; i += 128) {
        dst[i] = lds_pool[i] * sc[i >> 8];
    }
}

// ---------------------------------------------------------------------------
// Kernel B: barrier-free parallel cyclic Jacobi eigensolver, one matrix per
// wave (wave32). Cross-lane data flows through a per-wave LDS slice; CDNA5's
// LDS pipeline is in-order per wave, so only compiler scheduling fences
// (wave_barrier, no instruction emitted) + compiler dscnt waits are needed.
// 8 waves (8 matrices) per 256-thread block.
// ---------------------------------------------------------------------------
__global__ __launch_bounds__(256)
void mpff_eig_log_kernel(const float* __restrict__ pooled,
                         float* __restrict__ out)
{
    __shared__ float Aw[8][256];
    __shared__ float Vw[8][256];
    __shared__ float cow[8][16];
    __shared__ float cpw[8][16];
    __shared__ int   prtw[8][16];
    __shared__ float lvw[8][16];

    const int tid  = threadIdx.x;
    const int lane = tid & 31;
    const int wv   = tid >> 5;
    const size_t m = (size_t)blockIdx.x * 8 + wv;   // matrix index (grid covers exactly)

    float* A  = Aw[wv];
    float* V  = Vw[wv];
    float* co = cow[wv];
    float* cp = cpw[wv];
    int*   prt = prtw[wv];
    float* lv  = lvw[wv];

    const int jcol  = lane & 15;    // owned column
    const int ibase = lane >> 4;    // 0: even rows, 1: odd rows

    // load matrix, init V = I (lane owns elements e = lane + 32k)
    const float* src = pooled + m * 256;
    #pragma unroll
    for (int k = 0; k < 8; ++k) {
        int e = lane + 32 * k;
        A[e] = src[e];
        V[e] = ((e >> 4) == (e & 15)) ? 1.0f : 0.0f;
    }
    __builtin_amdgcn_wave_barrier();

    for (int sweep = 0; sweep < 10; ++sweep) {
        for (int r = 0; r < 15; ++r) {
            if (lane < 16) {
                // round-robin pairing: fix 15; i pairs with (2r - i) mod 15
                int ii = lane, pp;
                if (ii == 15) pp = r;
                else {
                    int j2 = (2 * r - ii) % 15;
                    if (j2 < 0) j2 += 15;
                    pp = (j2 == ii) ? 15 : j2;
                }
                prt[ii] = pp;
                if (ii < pp) {
                    float app = A[ii * 16 + ii];
                    float aqq = A[pp * 16 + pp];
                    float apq = A[ii * 16 + pp];
                    float c = 1.0f, s = 0.0f;
                    if (fabsf(apq) > 1e-36f) {
                        float tau = (aqq - app) / (2.0f * apq);
                        float tt  = ((tau >= 0.0f) ? 1.0f : -1.0f) /
                                    (fabsf(tau) + sqrtf(1.0f + tau * tau));
                        c = 1.0f / sqrtf(1.0f + tt * tt);
                        s = tt * c;
                    }
                    co[ii] = c; cp[ii] = -s;   // row p' = c*p - s*q
                    co[pp] = c; cp[pp] =  s;   // row q' = s*p + c*q
                }
            }
            __builtin_amdgcn_wave_barrier();

            const int   jp = prt[jcol];
            const float cj = co[jcol], pj = cp[jcol];
            float an[8], vn[8];
            #pragma unroll
            for (int k = 0; k < 8; ++k) {
                int i  = ibase + 2 * k;
                int ip = prt[i];
                float ci = co[i], pi = cp[i];
                float a00 = A[i * 16 + jcol],  a01 = A[i * 16 + jp];
                float a10 = A[ip * 16 + jcol], a11 = A[ip * 16 + jp];
                an[k] = cj * (ci * a00 + pi * a10) + pj * (ci * a01 + pi * a11);
                vn[k] = cj * V[i * 16 + jcol] + pj * V[i * 16 + jp];
            }
            __builtin_amdgcn_wave_barrier();
            #pragma unroll
            for (int k = 0; k < 8; ++k) {
                int i = ibase + 2 * k;
                A[i * 16 + jcol] = an[k];
                V[i * 16 + jcol] = vn[k];
            }
            __builtin_amdgcn_wave_barrier();
        }
    }

    // log of eigenvalues with nan/neginf -> 0 semantics
    if (lane < 16) {
        float l = A[lane * 16 + lane];
        lv[lane] = (l > 0.0f) ? logf(l) : 0.0f;
    }
    __builtin_amdgcn_wave_barrier();

    // reconstruct out[i][d] = sum_j V[i][j] * lv[j] * V[d][j] for owned elements
    float acc[8];
    #pragma unroll
    for (int k = 0; k < 8; ++k) acc[k] = 0.0f;
    #pragma unroll
    for (int j = 0; j < 16; ++j) {
        float vdj = V[jcol * 16 + j] * lv[j];
        #pragma unroll
        for (int k = 0; k < 8; ++k) {
            int i = ibase + 2 * k;
            acc[k] += V[i * 16 + j] * vdj;
        }
    }

    // out shape (b, kband, c, d, n): decode m = ((b*308)+n)*5 + kband
    const int b   = (int)(m / (NW * 5));
    const int rem = (int)(m % (NW * 5));
    const int n   = rem / 5;
    const int kb  = rem % 5;
    #pragma unroll
    for (int k = 0; k < 8; ++k) {
        int i = ibase + 2 * k;
        out[((((size_t)b * 5 + kb) * 16 + i) * 16 + jcol) * NW + n] = acc[k];
    }
}

extern "C" void kernel_launch(void* const* d_in, const int* in_sizes, int n_in,
                              void* d_out, int out_size, void* d_ws, size_t ws_size,
                              hipStream_t stream) {
    (void)in_sizes; (void)n_in; (void)out_size; (void)ws_size;
    const float* x   = (const float*)d_in[0];   // (16,16,20000) f32
    const float* win = (const float*)d_in[1];   // (64,) f32
    float* pooled = (float*)d_ws;               // 16*308*5*256 f32 = 25.2 MB
    float* outp   = (float*)d_out;              // (16,5,16,16,308) f32

    mpff_stft_csd_kernel<<<dim3(NB * NW), dim3(128), 0, stream>>>(x, win, pooled);
    // 24640 matrices, 8 per block (one per wave)
    mpff_eig_log_kernel<<<dim3((NB * NW * 5) / 8), dim3(256), 0, stream>>>(pooled, outp);
}